// LinearInterpolate_31198642438338
// MI455X (gfx1250) — compile-verified
//
#include <hip/hip_runtime.h>
#include <hip/hip_bf16.h>
#include <stdint.h>

// Problem shape (fixed by the reference)
#define TDIM 4096
#define BDIM 32
#define DDIM 128
#define NC   64                     // chunks per sequence
#define LCH  64                     // timesteps per chunk (NC*LCH == TDIM)
#define SEG  (BDIM * NC * DDIM)     // 262144 chunk-segments
#define TPB3 64                     // threads per block in final kernel

// ---------------- CDNA5 async global->LDS copy (builtin or inline asm) -------
#if defined(__has_builtin)
#  if __has_builtin(__builtin_amdgcn_global_load_async_to_lds_b32)
#    define HAVE_ASYNC_BUILTIN 1
#  endif
#  if __has_builtin(__builtin_amdgcn_s_wait_asynccnt)
#    define HAVE_WAITASYNC_BUILTIN 1
#  endif
#endif

typedef __attribute__((address_space(1))) int g_as_int;  // global
typedef __attribute__((address_space(3))) int l_as_int;  // LDS

__device__ __forceinline__ void async_copy_b32(const float* g, float* l) {
#if defined(HAVE_ASYNC_BUILTIN)
  __builtin_amdgcn_global_load_async_to_lds_b32(
      (g_as_int*)(void*)(g),
      (l_as_int*)(void*)(l),
      0, 0);
#else
  unsigned loff = (unsigned)(unsigned long long)(l_as_int*)(void*)(l);
  asm volatile("global_load_async_to_lds_b32 %0, %1, off"
               :: "v"(loff), "v"((unsigned long long)(g)) : "memory");
#endif
}

__device__ __forceinline__ void wait_async0() {
#if defined(HAVE_WAITASYNC_BUILTIN)
  __builtin_amdgcn_s_wait_asynccnt(0);
#else
  asm volatile("s_wait_asynccnt 0" ::: "memory");
#endif
}

// ---------------- Kernel 1: per-chunk first/last observation summaries -------
// ws[0*SEG+g]=x_first  ws[1*SEG+g]=t_first (3e38 sentinel = none)
// ws[2*SEG+g]=x_last   ws[3*SEG+g]=t_last  (-1   sentinel = none)
__global__ __launch_bounds__(256) void k1_summaries(
    const float* __restrict__ v, const float* __restrict__ t,
    const unsigned char* __restrict__ m, float* __restrict__ ws) {
  int g = blockIdx.x * 256 + threadIdx.x;        // segment id in [0, SEG)
  int d = g % DDIM;
  int c = (g / DDIM) % NC;
  int b = g / (DDIM * NC);
  int base = (b * TDIM + c * LCH) * DDIM + d;

  float xf = 0.0f, tf = 3.0e38f;
  float xl = 0.0f, tl = -1.0f;
  bool hasf = false;
#pragma unroll 4
  for (int i = 0; i < LCH; ++i) {
    int idx = base + i * DDIM;
    __builtin_prefetch(v + idx + 24 * DDIM, 0, 0);   // global_prefetch_b8
    __builtin_prefetch(t + idx + 24 * DDIM, 0, 0);
    unsigned char mk = m[idx];
    float vv = v[idx];
    float tt = t[idx];
    if (mk) {
      if (!hasf) { xf = vv; tf = tt; hasf = true; }
      xl = vv; tl = tt;
    }
  }
  ws[0 * SEG + g] = xf;
  ws[1 * SEG + g] = tf;
  ws[2 * SEG + g] = xl;
  ws[3 * SEG + g] = tl;
}

// ---------------- Kernel 2: scan chunk summaries into per-chunk carries ------
// ws[4*SEG+s]=carry x_last  ws[5*SEG+s]=carry t_last   (before chunk s)
// ws[6*SEG+s]=carry x_next  ws[7*SEG+s]=carry t_next   (after  chunk s)
// Edge fallbacks per reference: (0, times[b,0,d]) and (0, times[b,T-1,d]).
__global__ __launch_bounds__(256) void k2_carries(
    const float* __restrict__ t, float* __restrict__ ws) {
  int g = blockIdx.x * 256 + threadIdx.x;        // [0, B*D)
  if (g >= BDIM * DDIM) return;
  int d = g % DDIM;
  int b = g / DDIM;
  float t0 = t[b * TDIM * DDIM + d];
  float tE = t[b * TDIM * DDIM + (TDIM - 1) * DDIM + d];

  float cx = 0.0f, ct = t0;                      // forward carry
  for (int c = 0; c < NC; ++c) {
    int s = (b * NC + c) * DDIM + d;
    ws[4 * SEG + s] = cx;
    ws[5 * SEG + s] = ct;
    float tl = ws[3 * SEG + s];
    if (tl >= 0.0f) { cx = ws[2 * SEG + s]; ct = tl; }
  }
  cx = 0.0f; ct = tE;                            // backward carry
  for (int c = NC - 1; c >= 0; --c) {
    int s = (b * NC + c) * DDIM + d;
    ws[6 * SEG + s] = cx;
    ws[7 * SEG + s] = ct;
    float tf = ws[1 * SEG + s];
    if (tf < 1.0e37f) { cx = ws[0 * SEG + s]; ct = tf; }
  }
}

// ---------------- Kernel 3: async-stage tile, in-LDS bwd scan, fwd interp ----
__global__ __launch_bounds__(TPB3) void k3_fill(
    const float* __restrict__ v, const float* __restrict__ t,
    const unsigned char* __restrict__ m, const float* __restrict__ ws,
    float* __restrict__ out) {
  // 48KB static LDS; [i][tid] layout -> wave accesses 32 distinct banks
  __shared__ float s_xn[LCH][TPB3];  // staged value, overwritten with x_next
  __shared__ float s_ts[LCH][TPB3];  // staged time; sign bit encodes mask
  __shared__ float s_tn[LCH][TPB3];  // t_next

  int tid = threadIdx.x;
  int g = blockIdx.x * TPB3 + tid;               // segment id == ws index
  int d = g % DDIM;
  int c = (g / DDIM) % NC;
  int b = g / (DDIM * NC);
  int base = (b * TDIM + c * LCH) * DDIM + d;

  // Bulk async copy of this block's tile (values + times) into LDS.
  for (int i = 0; i < LCH; ++i) {
    async_copy_b32(v + base + i * DDIM, &s_xn[i][tid]);
    async_copy_b32(t + base + i * DDIM, &s_ts[i][tid]);
  }

  float xn = ws[6 * SEG + g];                    // next-obs carry-in
  float tn = ws[7 * SEG + g];
  wait_async0();

  // Backward in-chunk scan: rewrite tile in place with (x_next, t_next, +-t).
  // Thread-private LDS columns -> no barriers needed.
#pragma unroll 8
  for (int i = LCH - 1; i >= 0; --i) {
    unsigned char mk = m[base + i * DDIM];
    float vv = s_xn[i][tid];
    float tt = s_ts[i][tid];
    if (mk) { xn = vv; tn = tt; }                // inclusive next
    s_xn[i][tid] = xn;                           // observed: xn == value
    s_tn[i][tid] = tn;
    s_ts[i][tid] = mk ? -tt : tt;                // times > 0 -> sign is free
  }

  // Forward pass: only LDS reads + coalesced global store.
  float xl = ws[4 * SEG + g];                    // last-obs carry-in
  float tl = ws[5 * SEG + g];
#pragma unroll 8
  for (int i = 0; i < LCH; ++i) {
    float ts  = s_ts[i][tid];
    float xni = s_xn[i][tid];
    float tni = s_tn[i][tid];
    float r;
    if (ts < 0.0f) {                             // observed
      r = xni;                                   // == original value
      xl = xni; tl = tni;                        // == (value, time)
    } else {                                     // interpolate
      float den = tni - tl;
      float num = xl * (tni - ts) + xni * (ts - tl);
      r = (den != 0.0f) ? (num / den) : 0.0f;
    }
    out[base + i * DDIM] = r;
  }
}

// ---------------- Host-side launcher -----------------------------------------
extern "C" void kernel_launch(void* const* d_in, const int* in_sizes, int n_in,
                              void* d_out, int out_size, void* d_ws, size_t ws_size,
                              hipStream_t stream) {
  (void)in_sizes; (void)n_in; (void)out_size; (void)ws_size;
  const float*         v = (const float*)d_in[0];           // values  f32
  const float*         t = (const float*)d_in[1];           // times   f32
  const unsigned char* m = (const unsigned char*)d_in[2];   // mask    bool(1B)
  float* out = (float*)d_out;
  float* ws  = (float*)d_ws;   // 8 * SEG floats = 8 MB scratch

  k1_summaries<<<SEG / 256, 256, 0, stream>>>(v, t, m, ws);
  k2_carries<<<(BDIM * DDIM) / 256, 256, 0, stream>>>(t, ws);
  k3_fill<<<SEG / TPB3, TPB3, 0, stream>>>(v, t, m, ws, out);
}